// CrossBlock_56727928045699
// MI455X (gfx1250) — compile-verified
//
#include <hip/hip_runtime.h>
#include <hip/hip_bf16.h>
#include <math.h>

// ---------------------------------------------------------------------------
// SS2D (VMamba selective-scan 2D) for gfx1250 / MI455X.
// B=4, H=W=32, L=1024, DM=96, DI=192, K=4 dirs, N=16 state, R=6 dt-rank.
//  - all GEMMs: V_WMMA_F32_16X16X4_F32 (fp32 data, op is latency-bound)
//  - selective scan: chunked linear-recurrence (16 chunks x 64 steps),
//    3-phase (chunk transitions -> combine -> replay), B/C staged into LDS
//    with GLOBAL_LOAD_ASYNC_TO_LDS_B128 (ASYNCcnt) when available.
// ---------------------------------------------------------------------------

typedef __attribute__((ext_vector_type(2))) float v2f;
typedef __attribute__((ext_vector_type(8))) float v8f;
typedef int v4i __attribute__((vector_size(4 * sizeof(int))));  // matches builtin sig

#define WMMA_F32X4(a, b, c) \
  __builtin_amdgcn_wmma_f32_16x16x4_f32(false, (a), false, (b), (short)0, (c), false, false)

#if defined(__has_builtin)
#if __has_builtin(__builtin_amdgcn_global_load_async_to_lds_b128)
#define HAVE_ASYNC_LDS 1
#endif
#if __has_builtin(__builtin_amdgcn_s_wait_asynccnt)
#define HAVE_WAIT_ASYNC 1
#endif
#endif

static constexpr int Bb = 4;
static constexpr int HH = 32;
static constexpr int WW = 32;
static constexpr int LL = 1024;   // H*W
static constexpr int DM = 96;
static constexpr int DI = 192;
static constexpr int NS = 16;     // d_state
static constexpr int RD = 6;      // dt rank
// x_dbl row layout (padded to CP=52 floats = 208B, 16B-aligned rows):
//   cols [0,6)  : raw dt (rank-6)
//   cols [8,24) : B   (16 floats, starts 16B-aligned)
//   cols [24,40): C   (16 floats) -> B..C span = 32 contiguous floats = 128B
static constexpr int CP   = 52;
static constexpr int BOFF = 8;
static constexpr int NCH  = 16;   // scan chunks
static constexpr int CLEN = 64;   // steps per chunk (NCH*CLEN == LL)

// Direction index map (read row == merged write row; H==W so T is involution)
//  k=0: l   k=1: T(l)   k=2: L-1-l   k=3: T(L-1-l)
__device__ __forceinline__ int map_l(int k, int l) {
  int ll = (k & 2) ? (LL - 1 - l) : l;
  if (k & 1) ll = ((ll & 31) << 5) | (ll >> 5);
  return ll;
}

__device__ __forceinline__ void wait_asynccnt0() {
#if defined(HAVE_WAIT_ASYNC)
  __builtin_amdgcn_s_wait_asynccnt(0);
#elif defined(HAVE_ASYNC_LDS)
  asm volatile("s_wait_asynccnt 0x0" ::: "memory");
#endif
}

// Stage B||C (32 floats) for CLEN steps of chunk `ch` of (b,k) into sBC.
__device__ __forceinline__ void stage_bc(const float* __restrict__ xdbl, int bk,
                                         int ch, float* __restrict__ sBC) {
#if defined(HAVE_ASYNC_LDS)
  for (int i = threadIdx.x; i < CLEN * 8; i += DI) {
    const int r = i >> 3, s = i & 7;  // row in chunk, 16B segment
    const float* g = xdbl + (bk * LL + ch * CLEN + r) * CP + BOFF + s * 4;
    __builtin_amdgcn_global_load_async_to_lds_b128(
        (v4i*)(void*)g,
        (__attribute__((address_space(3))) v4i*)(
            __attribute__((address_space(3))) void*)(void*)&sBC[r * 32 + s * 4],
        0, 0);
  }
  wait_asynccnt0();
  __syncthreads();
#else
  for (int i = threadIdx.x; i < CLEN * 32; i += DI) {
    const int r = i >> 5, n = i & 31;
    sBC[i] = xdbl[(bk * LL + ch * CLEN + r) * CP + BOFF + n];
  }
  __syncthreads();
#endif
}

// ---------------------------------------------------------------------------
// 1) in_proj: xz = x @ W^T ; W (384,96). M=4096, N=384, K=96.
// ---------------------------------------------------------------------------
__global__ __launch_bounds__(128) void k_in_proj(const float* __restrict__ x,
                                                 const float* __restrict__ w,
                                                 float* __restrict__ xi,
                                                 float* __restrict__ z) {
  const int lane = threadIdx.x & 31;
  const int wv   = threadIdx.x >> 5;
  const int mt   = blockIdx.x;             // 0..255
  const int nt   = blockIdx.y * 4 + wv;    // 0..23
  const int half = lane >> 4;
  const int lm   = lane & 15;
  const int arow = mt * 16 + lm;
  const int brow = nt * 16 + lm;
  v8f acc = {};
  for (int k0 = 0; k0 < DM; k0 += 4) {
    const int col = k0 + half * 2;
    v2f a = *(const v2f*)(x + arow * DM + col);
    v2f b = *(const v2f*)(w + brow * DM + col);
    acc = WMMA_F32X4(a, b, acc);
  }
  const int ncol = nt * 16 + lm;
#pragma unroll
  for (int i = 0; i < 8; ++i) {
    const int r = mt * 16 + half * 8 + i;
    if (ncol < DI) xi[r * DI + ncol] = acc[i];
    else           z[r * DI + (ncol - DI)] = acc[i];
  }
}

// ---------------------------------------------------------------------------
// 2) depthwise 3x3 conv (SAME) + bias + SiLU.  xi (B,H,W,DI) -> xc (B,L,DI)
// ---------------------------------------------------------------------------
__global__ __launch_bounds__(192) void k_conv_silu(const float* __restrict__ xi,
                                                   const float* __restrict__ cw,
                                                   const float* __restrict__ cb,
                                                   float* __restrict__ xc) {
  const int pix = blockIdx.x;         // b*L + l
  const int d   = threadIdx.x;
  const int b   = pix >> 10;
  const int l   = pix & (LL - 1);
  const int h   = l >> 5, wq = l & 31;
  float acc = cb[d];
#pragma unroll
  for (int dy = -1; dy <= 1; ++dy) {
#pragma unroll
    for (int dx = -1; dx <= 1; ++dx) {
      const int hh = h + dy, ww = wq + dx;
      if (hh >= 0 && hh < HH && ww >= 0 && ww < WW) {
        const float v = xi[(b * LL + (hh * WW + ww)) * DI + d];
        acc += v * cw[d * 9 + (dy + 1) * 3 + (dx + 1)];
      }
    }
  }
  xc[(b * LL + l) * DI + d] = acc / (1.f + __expf(-acc));
}

// ---------------------------------------------------------------------------
// 3) x_dbl[b,k,l,c] = sum_d xs[b,k,d,l] * x_proj_weight[k,c,d]
//    GEMM per (b,k): M=1024, N=48 (pad of 38), K=192. Direction map folded
//    into A loads. Stored with the aligned CP=52 layout (dt|B|C).
// ---------------------------------------------------------------------------
__global__ __launch_bounds__(128) void k_xdbl(const float* __restrict__ xc,
                                              const float* __restrict__ xpw,
                                              float* __restrict__ xdbl) {
  const int lane = threadIdx.x & 31;
  const int wv   = threadIdx.x >> 5;
  const int mt   = blockIdx.x * 4 + wv;  // 0..63
  const int nt   = blockIdx.y;           // 0..2
  const int bk   = blockIdx.z;           // 0..15
  const int b    = bk >> 2, k = bk & 3;
  const int half = lane >> 4, lm = lane & 15;
  const int lrow = mt * 16 + lm;
  const int lmap = map_l(k, lrow);
  const float* Abase = xc + (b * LL + lmap) * DI;
  const int c = nt * 16 + lm;            // math col 0..47 (valid < 38)
  const float* Bbase = xpw + (k * 38 + c) * DI;
  const bool cvalid = (c < 38);
  v8f acc = {};
  for (int k0 = 0; k0 < DI; k0 += 4) {
    const int col = k0 + half * 2;
    v2f a = *(const v2f*)(Abase + col);
    v2f bf = {0.f, 0.f};
    if (cvalid) bf = *(const v2f*)(Bbase + col);
    acc = WMMA_F32X4(a, bf, acc);
  }
  const int sc = (c < RD) ? c : c + 2;   // dt: 0..5 ; B: 8..23 ; C: 24..39
#pragma unroll
  for (int i = 0; i < 8; ++i) {
    const int r = mt * 16 + half * 8 + i;
    xdbl[(bk * LL + r) * CP + sc] = acc[i];
  }
}

// ---------------------------------------------------------------------------
// 4) dts[b,k,l,d] = sum_r x_dbl[b,k,l,r] * dt_projs_weight[k,d,r]
//    GEMM per (b,k): M=1024, N=192, K=8 (pad of 6).
// ---------------------------------------------------------------------------
__global__ __launch_bounds__(128) void k_dtproj(const float* __restrict__ xdbl,
                                                const float* __restrict__ dtw,
                                                float* __restrict__ dts) {
  const int lane = threadIdx.x & 31;
  const int wv   = threadIdx.x >> 5;
  const int mt   = blockIdx.x;            // 0..63
  const int nt   = blockIdx.y * 4 + wv;   // 0..11
  const int bk   = blockIdx.z;            // 0..15
  const int k    = bk & 3;
  const int half = lane >> 4, lm = lane & 15;
  const int lrow = mt * 16 + lm;
  const int nrow = nt * 16 + lm;          // d index
  v8f acc = {};
#pragma unroll
  for (int k0 = 0; k0 < 8; k0 += 4) {
    const int col = k0 + half * 2;
    v2f a, bf;
    a[0]  = (col     < RD) ? xdbl[(bk * LL + lrow) * CP + col]     : 0.f;
    a[1]  = (col + 1 < RD) ? xdbl[(bk * LL + lrow) * CP + col + 1] : 0.f;
    bf[0] = (col     < RD) ? dtw[(k * DI + nrow) * RD + col]       : 0.f;
    bf[1] = (col + 1 < RD) ? dtw[(k * DI + nrow) * RD + col + 1]   : 0.f;
    acc = WMMA_F32X4(a, bf, acc);
  }
#pragma unroll
  for (int i = 0; i < 8; ++i) {
    const int r = mt * 16 + half * 8 + i;
    dts[(bk * LL + r) * DI + nrow] = acc[i];
  }
}

// ---------------------------------------------------------------------------
// 5a) scan pass 1: per chunk, compute transition (P = prod a, S = scan from 0)
//     for each of the 16 states. Grid (NCH, 16 bk), 192 threads (one d each).
// ---------------------------------------------------------------------------
__global__ __launch_bounds__(192) void k_scan_pass1(const float* __restrict__ xc,
                                                    const float* __restrict__ xdbl,
                                                    const float* __restrict__ dts,
                                                    const float* __restrict__ dtb,
                                                    const float* __restrict__ alogs,
                                                    float* __restrict__ pp,
                                                    float* __restrict__ ss) {
  const int ch = blockIdx.x;
  const int bk = blockIdx.y;
  const int b  = bk >> 2, k = bk & 3;
  const int d  = threadIdx.x;
  __shared__ float sBC[CLEN * 32];
  stage_bc(xdbl, bk, ch, sBC);

  float A[NS], P[NS], S[NS];
#pragma unroll
  for (int n = 0; n < NS; ++n) {
    A[n] = -__expf(alogs[(k * DI + d) * NS + n]);
    P[n] = 1.f;
    S[n] = 0.f;
  }
  const float bias = dtb[k * DI + d];

  for (int cl = 0; cl < CLEN; ++cl) {
    const int l  = ch * CLEN + cl;
    const int lm = map_l(k, l);
    const float xv = xc[(b * LL + lm) * DI + d];
    const float t  = dts[(bk * LL + l) * DI + d] + bias;
    const float delta = (t > 20.f) ? t : __logf(1.f + __expf(t));
    const float dx = delta * xv;
#pragma unroll
    for (int n = 0; n < NS; ++n) {
      const float a = __expf(delta * A[n]);
      S[n] = a * S[n] + dx * sBC[cl * 32 + n];
      P[n] = P[n] * a;
    }
  }
  const int base = ((bk * NCH + ch) * DI + d) * NS;
#pragma unroll
  for (int n = 0; n < NS; ++n) {
    pp[base + n] = P[n];
    ss[base + n] = S[n];
  }
}

// ---------------------------------------------------------------------------
// 5b) combine: h_start_{j+1} = P_j * h_start_j + S_j (serial over 16 chunks).
//     Grid 16 (bk), 192 threads (one d each).
// ---------------------------------------------------------------------------
__global__ __launch_bounds__(192) void k_scan_combine(const float* __restrict__ pp,
                                                      const float* __restrict__ ss,
                                                      float* __restrict__ hst) {
  const int bk = blockIdx.x;
  const int d  = threadIdx.x;
#pragma unroll
  for (int n = 0; n < NS; ++n) {
    float hs = 0.f;
    for (int j = 0; j < NCH; ++j) {
      const int idx = ((bk * NCH + j) * DI + d) * NS + n;
      hst[idx] = hs;
      hs = pp[idx] * hs + ss[idx];
    }
  }
}

// ---------------------------------------------------------------------------
// 5c) scan pass 2: replay chunk with correct h_start, emit y at the merged
//     position map_l(k,l). Grid (NCH, 16 bk), 192 threads.
// ---------------------------------------------------------------------------
__global__ __launch_bounds__(192) void k_scan_pass2(const float* __restrict__ xc,
                                                    const float* __restrict__ xdbl,
                                                    const float* __restrict__ dts,
                                                    const float* __restrict__ dtb,
                                                    const float* __restrict__ alogs,
                                                    const float* __restrict__ Dsv,
                                                    const float* __restrict__ hst,
                                                    float* __restrict__ ypart) {
  const int ch = blockIdx.x;
  const int bk = blockIdx.y;
  const int b  = bk >> 2, k = bk & 3;
  const int d  = threadIdx.x;
  __shared__ float sBC[CLEN * 32];
  stage_bc(xdbl, bk, ch, sBC);

  float A[NS], h[NS];
  const int base = ((bk * NCH + ch) * DI + d) * NS;
#pragma unroll
  for (int n = 0; n < NS; ++n) {
    A[n] = -__expf(alogs[(k * DI + d) * NS + n]);
    h[n] = hst[base + n];
  }
  const float bias = dtb[k * DI + d];
  const float Dd   = Dsv[k * DI + d];

  for (int cl = 0; cl < CLEN; ++cl) {
    const int l  = ch * CLEN + cl;
    const int lm = map_l(k, l);
    const float xv = xc[(b * LL + lm) * DI + d];
    const float t  = dts[(bk * LL + l) * DI + d] + bias;
    const float delta = (t > 20.f) ? t : __logf(1.f + __expf(t));
    const float dx = delta * xv;
    float y = 0.f;
#pragma unroll
    for (int n = 0; n < NS; ++n) {
      const float a = __expf(delta * A[n]);
      h[n] = a * h[n] + dx * sBC[cl * 32 + n];
      y += h[n] * sBC[cl * 32 + 16 + n];
    }
    ypart[(bk * LL + lm) * DI + d] = y + Dd * xv;
  }
}

// ---------------------------------------------------------------------------
// 6) merge 4 directions + LayerNorm(ch) + gate with SiLU(z) -> yg (B*L, 192)
// ---------------------------------------------------------------------------
__global__ __launch_bounds__(192) void k_merge_ln_gate(const float* __restrict__ ypart,
                                                       const float* __restrict__ zbuf,
                                                       const float* __restrict__ gamma,
                                                       const float* __restrict__ beta,
                                                       float* __restrict__ yg) {
  const int row = blockIdx.x;   // b*L + l
  const int d   = threadIdx.x;
  const int b   = row >> 10;
  const int l   = row & (LL - 1);
  float v = 0.f;
#pragma unroll
  for (int k = 0; k < 4; ++k)
    v += ypart[((b * 4 + k) * LL + l) * DI + d];

  __shared__ float sred[8];
  float s = v;
#pragma unroll
  for (int off = 16; off > 0; off >>= 1) s += __shfl_xor(s, off, 32);
  if ((threadIdx.x & 31) == 0) sred[threadIdx.x >> 5] = s;
  __syncthreads();
  if (threadIdx.x == 0) {
    float t = 0.f;
    for (int i = 0; i < 6; ++i) t += sred[i];
    sred[6] = t * (1.f / DI);
  }
  __syncthreads();
  const float mu = sred[6];
  const float dv = v - mu;
  float q = dv * dv;
  __syncthreads();
#pragma unroll
  for (int off = 16; off > 0; off >>= 1) q += __shfl_xor(q, off, 32);
  if ((threadIdx.x & 31) == 0) sred[threadIdx.x >> 5] = q;
  __syncthreads();
  if (threadIdx.x == 0) {
    float t = 0.f;
    for (int i = 0; i < 6; ++i) t += sred[i];
    sred[7] = t * (1.f / DI);
  }
  __syncthreads();
  const float var = sred[7];
  const float yn = dv * rsqrtf(var + 1e-5f) * gamma[d] + beta[d];
  const float zz = zbuf[row * DI + d];
  yg[row * DI + d] = yn * (zz / (1.f + __expf(-zz)));
}

// ---------------------------------------------------------------------------
// 7) out_proj: out = yg @ Wout^T ; Wout (96,192). M=4096, N=96, K=192.
// ---------------------------------------------------------------------------
__global__ __launch_bounds__(192) void k_out_proj(const float* __restrict__ yg,
                                                  const float* __restrict__ wo,
                                                  float* __restrict__ out) {
  const int lane = threadIdx.x & 31;
  const int nt   = threadIdx.x >> 5;     // 0..5
  const int mt   = blockIdx.x;           // 0..255
  const int half = lane >> 4, lm = lane & 15;
  const int arow = mt * 16 + lm;
  const int brow = nt * 16 + lm;
  v8f acc = {};
  for (int k0 = 0; k0 < DI; k0 += 4) {
    const int col = k0 + half * 2;
    v2f a  = *(const v2f*)(yg + arow * DI + col);
    v2f bf = *(const v2f*)(wo + brow * DI + col);
    acc = WMMA_F32X4(a, bf, acc);
  }
#pragma unroll
  for (int i = 0; i < 8; ++i) {
    const int r = mt * 16 + half * 8 + i;
    out[r * DM + nt * 16 + lm] = acc[i];
  }
}

// ---------------------------------------------------------------------------
extern "C" void kernel_launch(void* const* d_in, const int* in_sizes, int n_in,
                              void* d_out, int out_size, void* d_ws, size_t ws_size,
                              hipStream_t stream) {
  const float* x     = (const float*)d_in[0];   // (4,32,32,96)
  const float* wpin  = (const float*)d_in[1];   // (384,96)
  const float* cw    = (const float*)d_in[2];   // (192,1,3,3)
  const float* cb    = (const float*)d_in[3];   // (192,)
  const float* xpw   = (const float*)d_in[4];   // (4,38,192)
  const float* dtw   = (const float*)d_in[5];   // (4,192,6)
  const float* dtb   = (const float*)d_in[6];   // (4,192)
  const float* alogs = (const float*)d_in[7];   // (768,16)
  const float* Dsv   = (const float*)d_in[8];   // (768,)
  const float* gamma = (const float*)d_in[9];   // (192,)
  const float* beta  = (const float*)d_in[10];  // (192,)
  const float* wout  = (const float*)d_in[11];  // (96,192)
  float* out = (float*)d_out;                   // (4096,96)

  float* ws    = (float*)d_ws;
  float* xi    = ws;                       // 4096*192
  float* z     = xi    + Bb * LL * DI;
  float* xc    = z     + Bb * LL * DI;
  float* xdbl  = xc    + Bb * LL * DI;     // 16*1024*52
  float* dts   = xdbl  + Bb * 4 * LL * CP;
  float* ypart = dts   + Bb * 4 * LL * DI;
  float* yg    = ypart + Bb * 4 * LL * DI;
  float* pp    = yg    + Bb * LL * DI;     // 16*16*192*16
  float* ss    = pp    + Bb * 4 * NCH * DI * NS;
  float* hst   = ss    + Bb * 4 * NCH * DI * NS;

  k_in_proj<<<dim3(256, 6), 128, 0, stream>>>(x, wpin, xi, z);
  k_conv_silu<<<Bb * LL, DI, 0, stream>>>(xi, cw, cb, xc);
  k_xdbl<<<dim3(16, 3, 16), 128, 0, stream>>>(xc, xpw, xdbl);
  k_dtproj<<<dim3(64, 3, 16), 128, 0, stream>>>(xdbl, dtw, dts);
  k_scan_pass1<<<dim3(NCH, 16), DI, 0, stream>>>(xc, xdbl, dts, dtb, alogs, pp, ss);
  k_scan_combine<<<16, DI, 0, stream>>>(pp, ss, hst);
  k_scan_pass2<<<dim3(NCH, 16), DI, 0, stream>>>(xc, xdbl, dts, dtb, alogs, Dsv, hst, ypart);
  k_merge_ln_gate<<<Bb * LL, DI, 0, stream>>>(ypart, z, gamma, beta, yg);
  k_out_proj<<<256, DI, 0, stream>>>(yg, wout, out);
}